// ContrastivePhaseSelector_69063074120406
// MI455X (gfx1250) — compile-verified
//
#include <hip/hip_runtime.h>
#include <hip/hip_bf16.h>
#include <climits>

typedef float v2f __attribute__((ext_vector_type(2)));
typedef float v8f __attribute__((ext_vector_type(8)));

#define HH 512
#define WW 512
#define NC 5
#define NV 5
#define CCH 128
#define DD 64
#define NN 2048
#define TOPK 3
#define INV_TEMP (1.0f/0.07f)
#define LOG_HARD_W 0.6931471805599453f
#define NEG_INF_V (-1000000000.0f)

// ---- workspace layout (floats) ----
// [0..29]   peak coords: (c*3+k)*2 + {row, col}
// [32..46]  softmax weights w[c*3+k]
// [64..319] c_loss partials: block*2 + {sum, count}
#define WS_COORDS 0
#define WS_W      32
#define WS_PART   64

// ---- d_out layout (floats) ----
#define OUT_REW    0
#define OUT_LOGITS 5242880
#define OUT_CLOSS  5242895
#define OUT_CONS   5242896

// ------------------------------------------------------------------
// Kernel 1: 5x5 NMS + top-3 peaks for batch-0 maps (one block per class)
// ------------------------------------------------------------------
__global__ void peaks_kernel(const float* __restrict__ hm, float* __restrict__ ws) {
  const int c = blockIdx.x;
  const float* m = hm + (size_t)c * HH * WW;   // batch 0
  const int tid = threadIdx.x;

  float bv0 = -INFINITY, bv1 = -INFINITY, bv2 = -INFINITY;
  int   bi0 = INT_MAX,   bi1 = INT_MAX,   bi2 = INT_MAX;

  for (int p = tid; p < HH * WW; p += 256) {
    const int y = p >> 9, x = p & 511;
    const float v = m[p];
    float mx = -INFINITY;
    const int y0 = max(y - 2, 0), y1 = min(y + 2, HH - 1);
    const int x0 = max(x - 2, 0), x1 = min(x + 2, WW - 1);
    for (int yy = y0; yy <= y1; ++yy)
      for (int xx = x0; xx <= x1; ++xx)
        mx = fmaxf(mx, m[yy * WW + xx]);
    const float cand = (v == mx) ? v : 0.0f;  // NMS-suppressed -> 0 (matches hm*(hm==pool))
    // insert into sorted top-3, ties -> smaller index (jax top_k order)
    if (cand > bv2 || (cand == bv2 && p < bi2)) {
      bv2 = cand; bi2 = p;
      if (bv2 > bv1 || (bv2 == bv1 && bi2 < bi1)) {
        float tv = bv1; int ti = bi1; bv1 = bv2; bi1 = bi2; bv2 = tv; bi2 = ti;
        if (bv1 > bv0 || (bv1 == bv0 && bi1 < bi0)) {
          tv = bv0; ti = bi0; bv0 = bv1; bi0 = bi1; bv1 = tv; bi1 = ti;
        }
      }
    }
  }

  __shared__ float sv[768];
  __shared__ int   si[768];
  sv[tid * 3 + 0] = bv0; si[tid * 3 + 0] = bi0;
  sv[tid * 3 + 1] = bv1; si[tid * 3 + 1] = bi1;
  sv[tid * 3 + 2] = bv2; si[tid * 3 + 2] = bi2;
  __syncthreads();

  if (tid == 0) {
    for (int k = 0; k < TOPK; ++k) {
      float best = -INFINITY; int bidx = INT_MAX; int bslot = 0;
      for (int s = 0; s < 768; ++s) {
        if (si[s] < 0) continue;
        if (sv[s] > best || (sv[s] == best && si[s] < bidx)) {
          best = sv[s]; bidx = si[s]; bslot = s;
        }
      }
      si[bslot] = -1;
      ws[WS_COORDS + (c * TOPK + k) * 2 + 0] = (float)(bidx >> 9);  // row
      ws[WS_COORDS + (c * TOPK + k) * 2 + 1] = (float)(bidx & 511); // col
    }
  }
}

// ------------------------------------------------------------------
// Kernel 2: bilinear sample + 2-layer MLP + normalize + logits + softmax w
// one block, 64 threads (thread d owns output feature d)
// ------------------------------------------------------------------
__global__ void mlp_kernel(const float* __restrict__ fm,
                           const float* __restrict__ phase_embed,
                           const float* __restrict__ W1, const float* __restrict__ b1,
                           const float* __restrict__ W2, const float* __restrict__ b2,
                           const float* __restrict__ logit_scale,
                           float* __restrict__ ws, float* __restrict__ out) {
  const int d = threadIdx.x;
  __shared__ float s[CCH];
  __shared__ float h[DD];
  __shared__ float red[DD];
  __shared__ float logits[NV][TOPK];
  const float scale = fminf(__expf(logit_scale[0]), 100.0f);

  for (int v = 0; v < NV; ++v) {
    for (int k = 0; k < TOPK; ++k) {
      const float r  = ws[WS_COORDS + (v * TOPK + k) * 2 + 0];
      const float cl = ws[WS_COORDS + (v * TOPK + k) * 2 + 1];
      const float fy = floorf(r), fx = floorf(cl);
      const float wy = r - fy,    wx = cl - fx;
      const int y0 = min(max((int)fy, 0), HH - 1);
      const int y1 = min(y0 + 1, HH - 1);
      const int x0 = min(max((int)fx, 0), WW - 1);
      const int x1 = min(x0 + 1, WW - 1);
      for (int c = d; c < CCH; c += DD) {
        const float* f = fm + ((size_t)v * CCH + c) * HH * WW;
        const float v00 = f[y0 * WW + x0], v01 = f[y0 * WW + x1];
        const float v10 = f[y1 * WW + x0], v11 = f[y1 * WW + x1];
        s[c] = (v00 * (1.0f - wx) + v01 * wx) * (1.0f - wy) +
               (v10 * (1.0f - wx) + v11 * wx) * wy;
      }
      __syncthreads();
      float acc = b1[d];
      for (int c = 0; c < CCH; ++c) acc += s[c] * W1[c * DD + d];
      h[d] = fmaxf(acc, 0.0f);
      __syncthreads();
      float p = b2[d];
      for (int e = 0; e < DD; ++e) p += h[e] * W2[e * DD + d];
      red[d] = p * p;
      __syncthreads();
      for (int off = 32; off > 0; off >>= 1) {
        if (d < off) red[d] += red[d + off];
        __syncthreads();
      }
      const float nrm = fmaxf(sqrtf(red[0]), 1e-12f);
      __syncthreads();
      const float pe = p / nrm;
      red[d] = pe * phase_embed[d];
      __syncthreads();
      for (int off = 32; off > 0; off >>= 1) {
        if (d < off) red[d] += red[d + off];
        __syncthreads();
      }
      if (d == 0) logits[v][k] = red[0] * scale;
      __syncthreads();
    }
  }
  if (d < NV) {
    const float l0 = logits[d][0], l1 = logits[d][1], l2 = logits[d][2];
    const float mx = fmaxf(l0, fmaxf(l1, l2));
    const float e0 = __expf(l0 - mx), e1 = __expf(l1 - mx), e2 = __expf(l2 - mx);
    const float ssum = e0 + e1 + e2;
    ws[WS_W + d * TOPK + 0] = e0 / ssum;
    ws[WS_W + d * TOPK + 1] = e1 / ssum;
    ws[WS_W + d * TOPK + 2] = e2 / ssum;
    out[OUT_LOGITS + d * TOPK + 0] = l0;
    out[OUT_LOGITS + d * TOPK + 1] = l1;
    out[OUT_LOGITS + d * TOPK + 2] = l2;
  }
}

// ------------------------------------------------------------------
// Kernel 3: reweighted[b,c,0,h,w] = heatmap[b,c,h,w] + sum_k w[c,k]*gauss
// (bandwidth-bound: 21 MB read + 21 MB write)
// ------------------------------------------------------------------
__global__ void reweight_kernel(const float* __restrict__ hm,
                                const float* __restrict__ ws,
                                float* __restrict__ out) {
  const int idx = blockIdx.x * 256 + threadIdx.x;
  if (idx >= NC * HH * WW) return;
  const int c = idx >> 18;
  const int rem = idx & (HH * WW - 1);
  const int y = rem >> 9, x = rem & 511;
  float bias = 0.0f;
#pragma unroll
  for (int k = 0; k < TOPK; ++k) {
    const float pr = ws[WS_COORDS + (c * TOPK + k) * 2 + 0];
    const float pc = ws[WS_COORDS + (c * TOPK + k) * 2 + 1];
    const float dr = (float)y - pr;
    const float dc = (float)x - pc;
    bias += ws[WS_W + c * TOPK + k] * __expf(-(dr * dr + dc * dc) * (1.0f / 18.0f));
  }
#pragma unroll
  for (int b = 0; b < 4; ++b) {
    const size_t o = (size_t)(b * NC + c) * (HH * WW) + rem;
    out[OUT_REW + o] = hm[o] + bias;
  }
}

// ------------------------------------------------------------------
// Kernel 4: sim = (E @ E^T)/TEMP via V_WMMA_F32_16X16X4_F32, fused with
// masked streaming logsumexp (pos/neg) per row. 128 blocks x 128 threads.
// Block owns 16 rows (A block staged in LDS); each of 4 waves streams 2
// column tiles per iteration (two independent accumulator chains).
// ------------------------------------------------------------------
__global__ void sim_lse_kernel(const float* __restrict__ E,
                               const unsigned char* __restrict__ posM,
                               const unsigned char* __restrict__ negM,
                               const unsigned char* __restrict__ hardM,
                               float* __restrict__ partials) {
  const int rowbase = blockIdx.x * 16;
  const int tid  = threadIdx.x;
  const int wave = tid >> 5;
  const int lane = tid & 31;
  const int lhalf = lane >> 4;   // 0 or 1
  const int lmod  = lane & 15;

  // Stage A block (16 rows x 64 k) in LDS once; coalesced global loads.
  __shared__ float sA[16 * DD];   // 4 KB
  for (int i = tid; i < 16 * DD; i += 128)
    sA[i] = E[(size_t)rowbase * DD + i];
  __syncthreads();
  const float* aBase = &sA[lmod * DD + 2 * lhalf];  // this lane's A row slice

  // Per-lane streaming-LSE state: VGPR j of C holds row (j + 8*lhalf)
  float pm[8], ps[8], nm[8], ns[8];
  int hpmask = 0;  // bit j = row j saw a positive
#pragma unroll
  for (int j = 0; j < 8; ++j) { pm[j] = -INFINITY; ps[j] = 0.0f; nm[j] = -INFINITY; ns[j] = 0.0f; }

  // Wave w owns tile pairs {2w, 2w+1}, {2w+8, 2w+9}, ... -> 32 tiles total
  for (int ct = wave * 2; ct < NN / 16; ct += 8) {
    const int col0 = ct * 16 + lmod;
    const int col1 = col0 + 16;
    const float* Brow0 = E + (size_t)col0 * DD;   // column of sim = row of E
    const float* Brow1 = E + (size_t)col1 * DD;
    v8f acc0 = {};
    v8f acc1 = {};
#pragma unroll
    for (int ks = 0; ks < 16; ++ks) {
      v2f a, b0, b1;
      a.x  = aBase[ks * 4];          // contiguous pair -> ds_load_b64
      a.y  = aBase[ks * 4 + 1];
      b0.x = Brow0[ks * 4 + lhalf];
      b0.y = Brow0[ks * 4 + lhalf + 2];
      b1.x = Brow1[ks * 4 + lhalf];
      b1.y = Brow1[ks * 4 + lhalf + 2];
      acc0 = __builtin_amdgcn_wmma_f32_16x16x4_f32(
          false, a, false, b0, (short)0, acc0, false, false);
      acc1 = __builtin_amdgcn_wmma_f32_16x16x4_f32(
          false, a, false, b1, (short)0, acc1, false, false);
    }
#pragma unroll
    for (int t = 0; t < 2; ++t) {
      const int col = (t == 0) ? col0 : col1;
#pragma unroll
      for (int j = 0; j < 8; ++j) {
        const int r = rowbase + j + 8 * lhalf;
        const float val = ((t == 0) ? acc0[j] : acc1[j]) * INV_TEMP;
        const size_t midx = (size_t)r * NN + col;
        const bool p  = posM[midx]  != 0;
        const bool n  = negM[midx]  != 0;
        const bool hn = hardM[midx] != 0;
        const float xp = p ? val : NEG_INF_V;
        const float xn = (n ? val : NEG_INF_V) + (hn ? LOG_HARD_W : 0.0f);
        float mnew = fmaxf(pm[j], xp);
        ps[j] = ps[j] * __expf(pm[j] - mnew) + __expf(xp - mnew);
        pm[j] = mnew;
        mnew = fmaxf(nm[j], xn);
        ns[j] = ns[j] * __expf(nm[j] - mnew) + __expf(xn - mnew);
        nm[j] = mnew;
        hpmask |= p ? (1 << j) : 0;
      }
    }
  }

  // Cross-lane combine (deterministic order) via LDS
  __shared__ float sPM[16][64], sPS[16][64], sNM[16][64], sNS[16][64];
  __shared__ int   sFl[16][64];
  const int slot = wave * 16 + lmod;
#pragma unroll
  for (int j = 0; j < 8; ++j) {
    const int rl = j + 8 * lhalf;
    sPM[rl][slot] = pm[j]; sPS[rl][slot] = ps[j];
    sNM[rl][slot] = nm[j]; sNS[rl][slot] = ns[j];
    sFl[rl][slot] = (hpmask >> j) & 1;
  }
  __syncthreads();

  __shared__ float rowLoss[16];
  __shared__ int   rowValid[16];
  if (tid < 16) {
    float mp = -INFINITY, mn = -INFINITY; int fl = 0;
    for (int s2 = 0; s2 < 64; ++s2) {
      mp = fmaxf(mp, sPM[tid][s2]);
      mn = fmaxf(mn, sNM[tid][s2]);
      fl |= sFl[tid][s2];
    }
    float sp = 0.0f, sn = 0.0f;
    for (int s2 = 0; s2 < 64; ++s2) {
      sp += sPS[tid][s2] * __expf(sPM[tid][s2] - mp);
      sn += sNS[tid][s2] * __expf(sNM[tid][s2] - mn);
    }
    const float lsep = mp + __logf(sp);
    const float lsen = mn + __logf(sn);
    const float mx = fmaxf(lsep, lsen);
    const float logden = mx + __logf(__expf(lsep - mx) + __expf(lsen - mx));
    rowLoss[tid]  = fl ? (logden - lsep) : 0.0f;   // -(lse_pos - log_denom)
    rowValid[tid] = fl ? 1 : 0;
  }
  __syncthreads();
  if (tid == 0) {
    float s2 = 0.0f; int cnt = 0;
    for (int r = 0; r < 16; ++r) { s2 += rowLoss[r]; cnt += rowValid[r]; }
    partials[blockIdx.x * 2 + 0] = s2;
    partials[blockIdx.x * 2 + 1] = (float)cnt;
  }
}

// ------------------------------------------------------------------
// Kernel 5: temporal-consistency loss over adjacent phase_embeds
// ------------------------------------------------------------------
__global__ void cons_kernel(const float* __restrict__ E,
                            const float* __restrict__ nt,
                            float* __restrict__ out) {
  __shared__ float red[256];
  const int tid = threadIdx.x;
  float acc = 0.0f;
  for (int i = tid; i < NN - 1; i += 256) {
    float dot = 0.0f, na = 0.0f, nb = 0.0f;
    for (int d = 0; d < DD; ++d) {
      const float a = E[(size_t)i * DD + d];
      const float b = E[(size_t)(i + 1) * DD + d];
      dot += a * b; na += a * a; nb += b * b;
    }
    const float cosv = dot / fmaxf(sqrtf(na) * sqrtf(nb), 1e-8f);
    const float dx = nt[i * 3 + 0] - nt[(i + 1) * 3 + 0];
    const float dy = nt[i * 3 + 1] - nt[(i + 1) * 3 + 1];
    const float dz = nt[i * 3 + 2] - nt[(i + 1) * 3 + 2];
    const float td = sqrtf(dx * dx + dy * dy + dz * dz);
    const float same = (td < 0.1f) ? 1.0f : 0.0f;
    acc += same * fmaxf(0.9f * same - cosv, 0.0f);
  }
  red[tid] = acc;
  __syncthreads();
  for (int off = 128; off > 0; off >>= 1) {
    if (tid < off) red[tid] += red[tid + off];
    __syncthreads();
  }
  if (tid == 0) out[OUT_CONS] = red[0] / (float)(NN - 1);
}

// ------------------------------------------------------------------
// Kernel 6: finalize c_loss = sum(partials)/max(count,1)
// ------------------------------------------------------------------
__global__ void finalize_kernel(const float* __restrict__ partials, float* __restrict__ out) {
  if (threadIdx.x == 0 && blockIdx.x == 0) {
    float s = 0.0f, c = 0.0f;
    for (int i = 0; i < 128; ++i) { s += partials[i * 2]; c += partials[i * 2 + 1]; }
    out[OUT_CLOSS] = s / fmaxf(c, 1.0f);
  }
}

extern "C" void kernel_launch(void* const* d_in, const int* in_sizes, int n_in,
                              void* d_out, int out_size, void* d_ws, size_t ws_size,
                              hipStream_t stream) {
  const float* heatmap      = (const float*)d_in[0];
  const float* feature_map  = (const float*)d_in[1];
  const float* phase_embed  = (const float*)d_in[2];
  const float* W1           = (const float*)d_in[3];
  const float* b1           = (const float*)d_in[4];
  const float* W2           = (const float*)d_in[5];
  const float* b2           = (const float*)d_in[6];
  const float* logit_scale  = (const float*)d_in[7];
  const float* phase_embeds = (const float*)d_in[8];
  const float* next_targets = (const float*)d_in[9];
  const unsigned char* pos_mask  = (const unsigned char*)d_in[10];  // numpy bool = 1 byte
  const unsigned char* neg_mask  = (const unsigned char*)d_in[11];
  const unsigned char* hard_mask = (const unsigned char*)d_in[12];
  float* out = (float*)d_out;
  float* ws  = (float*)d_ws;

  (void)in_sizes; (void)n_in; (void)out_size; (void)ws_size;

  peaks_kernel<<<NC, 256, 0, stream>>>(heatmap, ws);
  mlp_kernel<<<1, 64, 0, stream>>>(feature_map, phase_embed, W1, b1, W2, b2,
                                   logit_scale, ws, out);
  reweight_kernel<<<(NC * HH * WW + 255) / 256, 256, 0, stream>>>(heatmap, ws, out);
  sim_lse_kernel<<<NN / 16, 128, 0, stream>>>(phase_embeds, pos_mask, neg_mask,
                                              hard_mask, ws + WS_PART);
  cons_kernel<<<1, 256, 0, stream>>>(phase_embeds, next_targets, out);
  finalize_kernel<<<1, 32, 0, stream>>>(ws + WS_PART, out);
}